// ViewGCNEncoder_50611894616713
// MI455X (gfx1250) — compile-verified
//
#include <hip/hip_runtime.h>
#include <hip/hip_bf16.h>

typedef __attribute__((ext_vector_type(2))) float v2f;
typedef __attribute__((ext_vector_type(8))) float v8f;

#define WMMA_F32(A, B, C)                                                  \
  __builtin_amdgcn_wmma_f32_16x16x4_f32(false, (A), false, (B), (short)0,  \
                                        (C), false, false)

// ---------------------------------------------------------------------------
// zero-fill (segment_sum accumulators must start at 0; harness poisons 0xAA)
// ---------------------------------------------------------------------------
__global__ void zero_f32(float* __restrict__ p, long long n) {
  long long i = (long long)blockIdx.x * blockDim.x + threadIdx.x;
  const long long stride = (long long)gridDim.x * blockDim.x;
  for (; i < n; i += stride) p[i] = 0.0f;
}

// ---------------------------------------------------------------------------
// out[N x H] = X[N x K] @ W[H x K]^T + b
// One wave32 computes a 16x64 strip (4 accumulators share each A-fragment).
// Software-pipelined: k+4 fragments are loaded before the k WMMAs issue, so
// the WMMAs wait only on the previous load batch (partial loadcnt), keeping
// the matrix pipe and the L2 load path overlapped.
//
// V_WMMA_F32_16X16X4_F32 layouts (ISA 7.12.2):
//   A (16x4): lanes 0-15 -> M=lane, K=k..k+1 ; lanes 16-31 -> M=lane-16,
//             K=k+2..k+3  => contiguous float2 per lane.
//   B (4x16): B[k][n] = W[n][k]; same half-split over K => contiguous float2.
//   D (16x16): VGPR r: lanes 0-15 -> (M=r, N=lane); lanes 16-31 -> (M=r+8).
// ---------------------------------------------------------------------------
__global__ void gemm_wmma_f32(const float* __restrict__ X,
                              const float* __restrict__ W,
                              const float* __restrict__ Bv,
                              float* __restrict__ out,
                              int N, int K, int H) {
  const int lane = threadIdx.x & 31;
  const int tile = blockIdx.x * (blockDim.x >> 5) + (threadIdx.x >> 5);
  const int colQuads = H >> 6;   // 64-wide column groups (H in {256,128,64})
  const int rowTiles = N >> 4;
  if (tile >= rowTiles * colQuads) return;  // wave-uniform: EXEC stays all-1s

  const int tm = tile / colQuads;
  const int tq = tile - tm * colQuads;
  const int l15 = lane & 15;
  const int hi = lane >> 4;      // 0 for lanes 0-15, 1 for lanes 16-31
  const int khalf = hi << 1;     // K sub-offset 0 or 2

  const float* __restrict__ xrow = X + (size_t)(tm * 16 + l15) * K + khalf;
  const float* __restrict__ wb = W + (size_t)(tq * 64 + l15) * K + khalf;
  const size_t wstep = (size_t)16 * K;   // next 16-column tile of W

  v8f acc0 = {}, acc1 = {}, acc2 = {}, acc3 = {};

  // Prologue: fragments for k = 0
  v2f a  = *(const v2f*)(xrow);
  v2f b0 = *(const v2f*)(wb);
  v2f b1 = *(const v2f*)(wb + wstep);
  v2f b2 = *(const v2f*)(wb + 2 * wstep);
  v2f b3 = *(const v2f*)(wb + 3 * wstep);

#pragma unroll 2
  for (int k = 4; k < K; k += 4) {
    // Issue next batch of loads before consuming the current fragments.
    const v2f a_n  = *(const v2f*)(xrow + k);
    const v2f b0_n = *(const v2f*)(wb + k);
    const v2f b1_n = *(const v2f*)(wb + wstep + k);
    const v2f b2_n = *(const v2f*)(wb + 2 * wstep + k);
    const v2f b3_n = *(const v2f*)(wb + 3 * wstep + k);
    acc0 = WMMA_F32(a, b0, acc0);
    acc1 = WMMA_F32(a, b1, acc1);
    acc2 = WMMA_F32(a, b2, acc2);
    acc3 = WMMA_F32(a, b3, acc3);
    a = a_n; b0 = b0_n; b1 = b1_n; b2 = b2_n; b3 = b3_n;
  }
  // Epilogue: last k-step
  acc0 = WMMA_F32(a, b0, acc0);
  acc1 = WMMA_F32(a, b1, acc1);
  acc2 = WMMA_F32(a, b2, acc2);
  acc3 = WMMA_F32(a, b3, acc3);

  const int colBase = tq * 64 + l15;
  const float bias0 = Bv[colBase];
  const float bias1 = Bv[colBase + 16];
  const float bias2 = Bv[colBase + 32];
  const float bias3 = Bv[colBase + 48];
  float* __restrict__ orow =
      out + (size_t)(tm * 16 + hi * 8) * H + colBase;
#pragma unroll
  for (int r = 0; r < 8; ++r) {
    orow[(size_t)r * H + 0]  = acc0[r] + bias0;
    orow[(size_t)r * H + 16] = acc1[r] + bias1;
    orow[(size_t)r * H + 32] = acc2[r] + bias2;
    orow[(size_t)r * H + 48] = acc3[r] + bias3;
  }
}

// ---------------------------------------------------------------------------
// SpMM scatter: out[rows[e]] += vals[e] * h[cols[e]]
// One wave per edge; edge index forced into an SGPR so rows/cols/vals become
// scalar (s_load) reads; lanes stripe features in float4 chunks and scatter
// with relaxed agent-scope fp32 atomics (global_atomic_add_f32 at L2).
// ---------------------------------------------------------------------------
__device__ __forceinline__ void atom_add(float* p, float v) {
  __hip_atomic_fetch_add(p, v, __ATOMIC_RELAXED, __HIP_MEMORY_SCOPE_AGENT);
}

__global__ void spmm_scatter(const int* __restrict__ rows,
                             const int* __restrict__ cols,
                             const float* __restrict__ vals,
                             const float* __restrict__ h,
                             float* __restrict__ out,
                             int E, int d) {
  const int lane = threadIdx.x & 31;
  const int e = __builtin_amdgcn_readfirstlane(
      blockIdx.x * (blockDim.x >> 5) + (threadIdx.x >> 5));
  if (e >= E) return;
  const int r = rows[e];
  const int c = cols[e];
  const float v = vals[e];
  const float* __restrict__ hsrc = h + (size_t)c * d;
  float* __restrict__ odst = out + (size_t)r * d;
  for (int f = lane * 4; f < d; f += 128) {
    const float4 hv = *(const float4*)(hsrc + f);
    atom_add(odst + f + 0, v * hv.x);
    atom_add(odst + f + 1, v * hv.y);
    atom_add(odst + f + 2, v * hv.z);
    atom_add(odst + f + 3, v * hv.w);
  }
}

// ---------------------------------------------------------------------------
// Fused leaky-ReLU (slope 0.2) + dropout mask (keep scale 1/(1-0.2) = 1.25)
// ---------------------------------------------------------------------------
__global__ void leaky_dropout(float* __restrict__ h,
                              const unsigned char* __restrict__ mask,
                              long long n) {
  long long i = (long long)blockIdx.x * blockDim.x + threadIdx.x;
  const long long stride = (long long)gridDim.x * blockDim.x;
  for (; i < n; i += stride) {
    float v = h[i];
    v = (v >= 0.0f) ? v : 0.2f * v;
    h[i] = mask[i] ? v * 1.25f : 0.0f;
  }
}

// ---------------------------------------------------------------------------
extern "C" void kernel_launch(void* const* d_in, const int* in_sizes, int n_in,
                              void* d_out, int out_size, void* d_ws, size_t ws_size,
                              hipStream_t stream) {
  const float* x  = (const float*)d_in[0];
  const int* rows = (const int*)d_in[1];
  const int* cols = (const int*)d_in[2];
  const float* vals = (const float*)d_in[3];
  const float* W1 = (const float*)d_in[4];
  const float* b1 = (const float*)d_in[5];
  const float* W2 = (const float*)d_in[6];
  const float* b2 = (const float*)d_in[7];
  const float* W3 = (const float*)d_in[8];
  const float* b3 = (const float*)d_in[9];
  const unsigned char* m1 = (const unsigned char*)d_in[10];
  const unsigned char* m2 = (const unsigned char*)d_in[11];

  const int IN = 256;
  const int N  = in_sizes[0] / IN;    // 50000
  const int E  = in_sizes[1];         // 800000
  const int H1 = in_sizes[5];         // 256
  const int H2 = in_sizes[7];         // 128
  const int HO = in_sizes[9];         // 64

  float* bufA = (float*)d_ws;                 // GEMM outputs (pre-aggregation)
  float* bufB = bufA + (size_t)N * H1;        // SpMM outputs / next-layer input
  float* zout = (float*)d_out;

  auto gemm = [&](const float* Xp, const float* Wp, const float* bp, float* Op,
                  int K, int H) {
    const int tiles = (N >> 4) * (H >> 6);    // 16x64 strips
    const int blocks = (tiles + 7) / 8;       // 8 waves (256 thr) per block
    gemm_wmma_f32<<<blocks, 256, 0, stream>>>(Xp, Wp, bp, Op, N, K, H);
  };
  auto spmm = [&](const float* hp, float* op, int d) {
    const int blocks = (E + 7) / 8;           // wave per edge, 8 waves/block
    spmm_scatter<<<blocks, 256, 0, stream>>>(rows, cols, vals, hp, op, E, d);
  };
  auto zero = [&](float* p, long long n) {
    int blocks = (int)((n + 255) / 256);
    if (blocks > 65536) blocks = 65536;       // grid-stride inside
    zero_f32<<<blocks, 256, 0, stream>>>(p, n);
  };
  auto act = [&](float* p, const unsigned char* m, long long n) {
    int blocks = (int)((n + 255) / 256);
    if (blocks > 65536) blocks = 65536;
    leaky_dropout<<<blocks, 256, 0, stream>>>(p, m, n);
  };

  // ---- Layer 1: h = spmm(x@W1^T + b1); leaky; dropout(mask1) ----
  gemm(x, W1, b1, bufA, IN, H1);
  zero(bufB, (long long)N * H1);
  spmm(bufA, bufB, H1);
  act(bufB, m1, (long long)N * H1);

  // ---- Layer 2: h = spmm(h@W2^T + b2); leaky; dropout(mask2) ----
  gemm(bufB, W2, b2, bufA, H1, H2);
  zero(bufB, (long long)N * H2);
  spmm(bufA, bufB, H2);
  act(bufB, m2, (long long)N * H2);

  // ---- Layer 3: z = spmm(h@W3^T + b3) -> d_out ----
  gemm(bufB, W3, b3, bufA, H2, HO);
  zero(zout, (long long)N * HO);
  spmm(bufA, zout, HO);
}